// QuantumAttention_65481071401576
// MI455X (gfx1250) — compile-verified
//
#include <hip/hip_runtime.h>
#include <hip/hip_bf16.h>

#define NQ      8
#define DIM     256          // 2^NQ
#define BATCH   8192
#define NLAYERS 2

#define ROWS_PER_BLOCK  32   // batch rows per block (two 16-row WMMA blocks)
#define WAVES_PER_BLOCK 4    // each wave owns 4 of the 16 j-tiles
#define JT_PER_WAVE     (16 / WAVES_PER_BLOCK)
#define BLOCK_THREADS   (32 * WAVES_PER_BLOCK)

typedef __attribute__((ext_vector_type(2))) float v2f;
typedef __attribute__((ext_vector_type(8))) float v8f;

struct cplx { float re, im; };
__device__ __forceinline__ cplx cmul(cplx a, cplx b) {
    return { a.re * b.re - a.im * b.im, a.re * b.im + a.im * b.re };
}
__device__ __forceinline__ cplx cadd(cplx a, cplx b) { return { a.re + b.re, a.im + b.im }; }

// --------------------------------------------------------------------------
// Kernel 1: build the shared variational unitary, stored TRANSPOSED k-major:
//   Ut[k*256 + j] = U[j][k]   (so B fragments load coalesced in kernel 2)
// 16 columns of the identity per block evolved in LDS. Wire i <-> bit (7-i).
// --------------------------------------------------------------------------
#define K1_COLS 16

__global__ void build_unitary(const float* __restrict__ params,
                              float* __restrict__ Utre,
                              float* __restrict__ Utim) {
    __shared__ float sRe[K1_COLS * DIM];
    __shared__ float sIm[K1_COLS * DIM];
    const int tid = threadIdx.x;
    const int colBase = blockIdx.x * K1_COLS;   // this block's k range

    for (int idx = tid; idx < K1_COLS * DIM; idx += blockDim.x) {
        int col = idx >> 8, row = idx & 255;
        sRe[idx] = (row == colBase + col) ? 1.0f : 0.0f;
        sIm[idx] = 0.0f;
    }
    __syncthreads();

    auto apply1q = [&](cplx g00, cplx g01, cplx g10, cplx g11, int wire) {
        const int stride = 1 << (7 - wire);
        for (int t = tid; t < K1_COLS * (DIM / 2); t += blockDim.x) {
            int col = t >> 7, pr = t & 127;
            int low = pr & (stride - 1);
            int r0  = ((pr ^ low) << 1) | low;
            int r1  = r0 | stride;
            int i0 = col * DIM + r0, i1 = col * DIM + r1;
            cplx a0 = { sRe[i0], sIm[i0] };
            cplx a1 = { sRe[i1], sIm[i1] };
            cplx n0 = cadd(cmul(g00, a0), cmul(g01, a1));
            cplx n1 = cadd(cmul(g10, a0), cmul(g11, a1));
            sRe[i0] = n0.re; sIm[i0] = n0.im;
            sRe[i1] = n1.re; sIm[i1] = n1.im;
        }
        __syncthreads();
    };

    auto cnot_adj = [&](int ctrl) {     // target = ctrl + 1 (adjacent bits)
        const int pt = 6 - ctrl;        // target bit position; ctrl bit = pt+1
        const int ts = 1 << pt;
        for (int t = tid; t < K1_COLS * (DIM / 4); t += blockDim.x) {
            int col = t >> 6, pr = t & 63;
            int low = pr & (ts - 1);
            int r0 = ((pr >> pt) << (pt + 2)) | (2 << pt) | low;  // ctrl=1, tgt=0
            int r1 = r0 | ts;                                     // ctrl=1, tgt=1
            int i0 = col * DIM + r0, i1 = col * DIM + r1;
            float tr = sRe[i0]; sRe[i0] = sRe[i1]; sRe[i1] = tr;
            float ti = sIm[i0]; sIm[i0] = sIm[i1]; sIm[i1] = ti;
        }
        __syncthreads();
    };

    for (int l = 0; l < NLAYERS; ++l) {
        for (int i = 0; i < NQ; ++i) {
            float c, s;
            sincosf(0.5f * params[(l * NQ + i) * 3 + 0], &s, &c);
            apply1q({ c, 0 }, { -s, 0 }, { s, 0 }, { c, 0 }, i);      // RY
            sincosf(0.5f * params[(l * NQ + i) * 3 + 1], &s, &c);
            apply1q({ c, -s }, { 0, 0 }, { 0, 0 }, { c, s }, i);      // RZ
            if (i < NQ - 1) cnot_adj(i);
        }
        for (int i = 0; i < NQ; ++i) {
            float c, s;
            sincosf(0.5f * params[(l * NQ + i) * 3 + 2], &s, &c);
            apply1q({ c, 0 }, { 0, -s }, { 0, -s }, { c, 0 }, i);     // RX
        }
    }

    // transposed (k-major) write-out, coalesced over j
    for (int idx = tid; idx < K1_COLS * DIM; idx += blockDim.x) {
        int col = idx >> 8, j = idx & 255;
        Utre[(colBase + col) * DIM + j] = sRe[col * DIM + j];
        Utim[(colBase + col) * DIM + j] = sIm[col * DIM + j];
    }
}

// --------------------------------------------------------------------------
// Kernel 2: 128 threads (4 waves); 32 batch rows per block.
//   - shared S tile built closed-form by all 4 waves into dynamic LDS
//   - each wave owns 4 j-tiles; B fragments (k-major U^T) are coalesced
//     b32 loads off a hoisted per-lane base with immediate offsets
//   - 8 x V_WMMA_F32_16X16X4_F32 per k-step (complex MMA, 2 row blocks)
//   - deterministic cross-wave <Z> reduction through LDS, direct Out store
// --------------------------------------------------------------------------
#define PITCH 260   // row stride % 64 == 4 -> conflict-free ds_load_b64

__global__ void qgemm_wmma(const float* __restrict__ X,
                           const float* __restrict__ Utre,
                           const float* __restrict__ Utim,
                           float* __restrict__ Out) {
    extern __shared__ float lds[];
    float* sReT    = lds;                                   // 32 * PITCH
    float* sImT    = sReT + ROWS_PER_BLOCK * PITCH;
    float* csC     = sImT + ROWS_PER_BLOCK * PITCH;         // 32 * 8
    float* csS     = csC + ROWS_PER_BLOCK * NQ;
    float* partBuf = csS + ROWS_PER_BLOCK * NQ;             // 4 * 32 * 8

    const int tid  = threadIdx.x;
    const int lane = tid & 31;
    const int wave = tid >> 5;
    const int base = blockIdx.x * ROWS_PER_BLOCK;

    // per-row cos/sin of the encoding angles (all 4 waves cooperate)
    for (int t = tid; t < ROWS_PER_BLOCK * NQ; t += BLOCK_THREADS) {
        int r = t >> 3, i = t & 7;
        float s, c;
        sincosf(0.5f * X[(base + r) * NQ + i], &s, &c);
        csC[r * NQ + i] = c;
        csS[r * NQ + i] = s;
    }
    __syncthreads();

    // closed-form data-encoded state tile: amp = m * (-i)^popcount(k)
    for (int t = tid; t < ROWS_PER_BLOCK * DIM; t += BLOCK_THREADS) {
        int r = t >> 8, k = t & 255;
        float m = 1.0f;
#pragma unroll
        for (int i = 0; i < NQ; ++i)
            m *= ((k >> (7 - i)) & 1) ? csS[r * NQ + i] : csC[r * NQ + i];
        int p = __popc(k) & 3;
        sReT[r * PITCH + k] = (p == 0) ? m : ((p == 2) ? -m : 0.0f);
        sImT[r * PITCH + k] = (p == 3) ? m : ((p == 1) ? -m : 0.0f);
    }
    __syncthreads();

    const int half = lane >> 4;          // 0: lanes 0-15, 1: lanes 16-31
    const int mrow = lane & 15;
    const int kOff = half << 1;          // K sub-offset {0,2} per ISA A/B layout

    float zacc[2][8][8];
#pragma unroll
    for (int b = 0; b < 2; ++b)
#pragma unroll
        for (int r = 0; r < 8; ++r)
#pragma unroll
            for (int i = 0; i < 8; ++i) zacc[b][r][i] = 0.0f;

    const int jtBase = wave * JT_PER_WAVE;
    for (int jc = 0; jc < JT_PER_WAVE; ++jc) {
        const int jt = jtBase + jc;
        const int j  = jt * 16 + mrow;                  // this lane's output col
        // per-lane base into k-major U^T; loop offsets are uniform immediates
        const float* bre = Utre + (unsigned)(kOff * DIM) + j;
        const float* bim = Utim + (unsigned)(kOff * DIM) + j;
        __builtin_prefetch(bre, 0, 2);
        __builtin_prefetch(bim, 0, 2);

        v8f cre0 = {}, cim0 = {}, cre1 = {}, cim1 = {};
        for (int kk = 0; kk < DIM; kk += 4) {
            const unsigned ko = (unsigned)kk * DIM;     // constant-step offset
            // B fragment (coalesced b32 loads), shared by both row blocks
            v2f bRe = { bre[ko], bre[ko + DIM] };
            v2f bIm = { bim[ko], bim[ko + DIM] };
            // A fragments for rows [0..15] and [16..31] of this block
            const unsigned ka = (unsigned)(kk + kOff);
            const float2 aRe20 = *reinterpret_cast<const float2*>(&sReT[mrow * PITCH + ka]);
            const float2 aIm20 = *reinterpret_cast<const float2*>(&sImT[mrow * PITCH + ka]);
            const float2 aRe21 = *reinterpret_cast<const float2*>(&sReT[(mrow + 16) * PITCH + ka]);
            const float2 aIm21 = *reinterpret_cast<const float2*>(&sImT[(mrow + 16) * PITCH + ka]);
            v2f aRe0  = { aRe20.x,  aRe20.y };
            v2f aIm0  = { aIm20.x,  aIm20.y };
            v2f aImN0 = { -aIm20.x, -aIm20.y };  // f32 WMMA has no A-neg modifier
            v2f aRe1  = { aRe21.x,  aRe21.y };
            v2f aIm1  = { aIm21.x,  aIm21.y };
            v2f aImN1 = { -aIm21.x, -aIm21.y };
            // psi_re += aRe*bRe - aIm*bIm ; psi_im += aRe*bIm + aIm*bRe
            cre0 = __builtin_amdgcn_wmma_f32_16x16x4_f32(false, aRe0,  false, bRe, (short)0, cre0, false, false);
            cre0 = __builtin_amdgcn_wmma_f32_16x16x4_f32(false, aImN0, false, bIm, (short)0, cre0, false, false);
            cim0 = __builtin_amdgcn_wmma_f32_16x16x4_f32(false, aRe0,  false, bIm, (short)0, cim0, false, false);
            cim0 = __builtin_amdgcn_wmma_f32_16x16x4_f32(false, aIm0,  false, bRe, (short)0, cim0, false, false);
            cre1 = __builtin_amdgcn_wmma_f32_16x16x4_f32(false, aRe1,  false, bRe, (short)0, cre1, false, false);
            cre1 = __builtin_amdgcn_wmma_f32_16x16x4_f32(false, aImN1, false, bIm, (short)0, cre1, false, false);
            cim1 = __builtin_amdgcn_wmma_f32_16x16x4_f32(false, aRe1,  false, bIm, (short)0, cim1, false, false);
            cim1 = __builtin_amdgcn_wmma_f32_16x16x4_f32(false, aIm1,  false, bRe, (short)0, cim1, false, false);
        }
        // signed |psi|^2 accumulation; this lane's j is fixed within the tile
        float sgn[8];
#pragma unroll
        for (int i = 0; i < 8; ++i)
            sgn[i] = ((j >> (7 - i)) & 1) ? -1.0f : 1.0f;
#pragma unroll
        for (int r = 0; r < 8; ++r) {
            float p0 = cre0[r] * cre0[r] + cim0[r] * cim0[r];
            float p1 = cre1[r] * cre1[r] + cim1[r] * cim1[r];
#pragma unroll
            for (int i = 0; i < 8; ++i) {
                zacc[0][r][i] += sgn[i] * p0;
                zacc[1][r][i] += sgn[i] * p1;
            }
        }
    }

    // in-half lane reduction (xor masks 1,2,4,8 stay within each 16-lane half),
    // then stash this wave's per-row partials in LDS
#pragma unroll
    for (int b = 0; b < 2; ++b) {
#pragma unroll
        for (int r = 0; r < 8; ++r) {
#pragma unroll
            for (int i = 0; i < 8; ++i) {
                float v = zacc[b][r][i];
                v += __shfl_xor(v, 1);
                v += __shfl_xor(v, 2);
                v += __shfl_xor(v, 4);
                v += __shfl_xor(v, 8);
                if (mrow == 0) {
                    int row = b * 16 + r + half * 8;
                    partBuf[wave * (ROWS_PER_BLOCK * NQ) + row * NQ + i] = v;
                }
            }
        }
    }
    __syncthreads();

    // deterministic cross-wave sum and direct output store
    for (int t = tid; t < ROWS_PER_BLOCK * NQ; t += BLOCK_THREADS) {
        float v = partBuf[t]
                + partBuf[1 * (ROWS_PER_BLOCK * NQ) + t]
                + partBuf[2 * (ROWS_PER_BLOCK * NQ) + t]
                + partBuf[3 * (ROWS_PER_BLOCK * NQ) + t];
        Out[base * NQ + t] = v;
    }
}

// --------------------------------------------------------------------------
extern "C" void kernel_launch(void* const* d_in, const int* in_sizes, int n_in,
                              void* d_out, int out_size, void* d_ws, size_t ws_size,
                              hipStream_t stream) {
    const float* x      = (const float*)d_in[0];   // (8192, 8) f32
    const float* params = (const float*)d_in[1];   // (2, 8, 3) f32
    float* Utre = (float*)d_ws;                    // 256*256 f32 (k-major U^T)
    float* Utim = Utre + DIM * DIM;                // 256*256 f32

    build_unitary<<<DIM / K1_COLS, 256, 0, stream>>>(params, Utre, Utim);

    size_t ldsBytes = (size_t)(2 * ROWS_PER_BLOCK * PITCH +   // S tiles
                               2 * ROWS_PER_BLOCK * NQ +      // cos/sin
                               WAVES_PER_BLOCK * ROWS_PER_BLOCK * NQ)  // partials
                      * sizeof(float);
    qgemm_wmma<<<BATCH / ROWS_PER_BLOCK, BLOCK_THREADS, ldsBytes, stream>>>(
        x, Utre, Utim, (float*)d_out);
}